// GraphGATClassifier_1949915152974
// MI455X (gfx1250) — compile-verified
//
#include <hip/hip_runtime.h>
#include <hip/hip_bf16.h>
#include <math.h>

typedef float v2f __attribute__((ext_vector_type(2)));
typedef float v8f __attribute__((ext_vector_type(8)));

#define HC 128   // H*C
#define Hh 4
#define Cc 32
#define ED 16

// ---------- helpers ----------
__device__ __forceinline__ unsigned fenc(float f) {
    unsigned u = __float_as_uint(f);
    return (u & 0x80000000u) ? ~u : (u | 0x80000000u);
}
__device__ __forceinline__ float fdec(unsigned u) {
    return __uint_as_float((u & 0x80000000u) ? (u & 0x7FFFFFFFu) : ~u);
}

__global__ void zero_f32(float* p, long long n) {
    long long i = (long long)blockIdx.x * blockDim.x + threadIdx.x;
    if (i < n) p[i] = 0.0f;
}

// ---------- WMMA GEMM: Y[n,128] = X[n,128] @ W[128,128] ----------
// block = 256 (8 waves); each wave owns a 16-row strip, all 8 col tiles.
// Out-of-range rows are CLAMPED to nrows-1 (their D rows are dropped at the
// guarded store), so the A fetch is a single unpredicated global_load_b64.
__global__ __launch_bounds__(256) void gemm128_wmma(
    const float* __restrict__ X, const float* __restrict__ W,
    float* __restrict__ Y, int nrows) {
    __shared__ float Ws[HC * HC];           // 64 KB
    for (int i = threadIdx.x; i < HC * HC; i += 256) Ws[i] = W[i];
    __syncthreads();

    int wave = threadIdx.x >> 5;
    int lane = threadIdx.x & 31;
    int half = lane >> 4;                   // 0: K 0..1, 1: K 2..3
    int m    = lane & 15;
    int row0 = blockIdx.x * 128 + wave * 16;
    if (row0 >= nrows) return;
    int row = row0 + m;
    if (row >= nrows) row = nrows - 1;      // clamp: no exec-mask ladder
    const float* xr = X + (size_t)row * HC + 2 * half;

    v8f acc[8] = {};
    for (int k = 0; k < HC; k += 4) {
        v2f a = *(const v2f*)(xr + k);      // one global_load_b64
#pragma unroll
        for (int t = 0; t < 8; ++t) {
            int n = t * 16 + m;
            v2f b;
            b.x = Ws[(k + half * 2 + 0) * HC + n];
            b.y = Ws[(k + half * 2 + 1) * HC + n];
            acc[t] = __builtin_amdgcn_wmma_f32_16x16x4_f32(
                false, a, false, b, (short)0, acc[t], false, false);
        }
    }
#pragma unroll
    for (int t = 0; t < 8; ++t)
#pragma unroll
        for (int v = 0; v < 8; ++v) {
            int r = row0 + v + 8 * half;    // C/D layout: VGPR v -> row v / v+8
            if (r < nrows) Y[(size_t)r * HC + t * 16 + m] = acc[t][v];
        }
}

// ---------- per-node attention dots ----------
__global__ void att_dots(const float* __restrict__ xw,
                         const float* __restrict__ att_s,
                         const float* __restrict__ att_d,
                         float* __restrict__ asrc, float* __restrict__ adst, int n) {
    int i = blockIdx.x * blockDim.x + threadIdx.x;
    if (i >= n * Hh) return;
    int node = i >> 2, h = i & 3;
    const float* xr = xw + (size_t)node * HC + h * Cc;
    float s = 0.0f, d = 0.0f;
#pragma unroll
    for (int c = 0; c < Cc; ++c) { s += xr[c] * att_s[h * Cc + c]; d += xr[c] * att_d[h * Cc + c]; }
    asrc[i] = s; adst[i] = d;
}

// ---------- edge_attr column sums (into acc16) ----------
__global__ void edge_attr_sum(const float* __restrict__ ea, float* __restrict__ acc16, int E_) {
    __shared__ float red[256];
    float loc[ED];
#pragma unroll
    for (int d = 0; d < ED; ++d) loc[d] = 0.0f;
    for (int r = blockIdx.x * blockDim.x + threadIdx.x; r < E_; r += gridDim.x * blockDim.x) {
        const float* row = ea + (size_t)r * ED;
#pragma unroll
        for (int d = 0; d < ED; ++d) loc[d] += row[d];
    }
    for (int d = 0; d < ED; ++d) {
        red[threadIdx.x] = loc[d];
        __syncthreads();
        for (int off = 128; off; off >>= 1) {
            if (threadIdx.x < off) red[threadIdx.x] += red[threadIdx.x + off];
            __syncthreads();
        }
        if (threadIdx.x == 0) atomicAdd(&acc16[d], red[0]);
        __syncthreads();
    }
}

// ---------- fold att_e into W_e: M[d,h] = sum_c W_e[d, h*32+c]*att_e[h,c] ----------
__global__ void fold_We(const float* __restrict__ We1, const float* __restrict__ ae1,
                        const float* __restrict__ We2, const float* __restrict__ ae2,
                        float* __restrict__ M1, float* __restrict__ M2) {
    int i = threadIdx.x;
    if (i >= ED * Hh) return;
    int d = i >> 2, h = i & 3;
    float s1 = 0.0f, s2 = 0.0f;
#pragma unroll
    for (int c = 0; c < Cc; ++c) {
        s1 += We1[d * HC + h * Cc + c] * ae1[h * Cc + c];
        s2 += We2[d * HC + h * Cc + c] * ae2[h * Cc + c];
    }
    M1[i] = s1; M2[i] = s2;
}

// ---------- self-loop edge-attention constants: loop[h] = mean_ea @ M[:,h] ----------
__global__ void loop_consts(const float* __restrict__ acc16,
                            const float* __restrict__ M1, const float* __restrict__ M2,
                            float* __restrict__ L1, float* __restrict__ L2, float invE) {
    int h = threadIdx.x;
    if (h >= Hh) return;
    float s1 = 0.0f, s2 = 0.0f;
    for (int d = 0; d < ED; ++d) {
        float m = acc16[d] * invE;
        s1 += m * M1[d * Hh + h];
        s2 += m * M2[d * Hh + h];
    }
    L1[h] = s1; L2[h] = s2;
}

// ---------- per-edge edge-attention logits for both layers ----------
__global__ void edge_logits(const float* __restrict__ ea,
                            const float* __restrict__ M1, const float* __restrict__ M2,
                            float* __restrict__ ae1, float* __restrict__ ae2, int E_) {
    int e = blockIdx.x * blockDim.x + threadIdx.x;
    if (e >= E_) return;
    float row[ED];
    const float* er = ea + (size_t)e * ED;
#pragma unroll
    for (int d = 0; d < ED; ++d) row[d] = er[d];
#pragma unroll
    for (int h = 0; h < Hh; ++h) {
        float s1 = 0.0f, s2 = 0.0f;
#pragma unroll
        for (int d = 0; d < ED; ++d) { s1 += row[d] * M1[d * Hh + h]; s2 += row[d] * M2[d * Hh + h]; }
        ae1[(size_t)e * Hh + h] = s1;
        ae2[(size_t)e * Hh + h] = s2;
    }
}

// ---------- pass A: segment max of leaky-relu logits ----------
__global__ void edge_passA(const int* __restrict__ ei,
                           const float* __restrict__ ae, const float* __restrict__ loopc,
                           const float* __restrict__ asrc, const float* __restrict__ adst,
                           unsigned* __restrict__ amax, int E_, int N_) {
    int e = blockIdx.x * blockDim.x + threadIdx.x;
    if (e >= E_ + N_) return;
    int s, d; float a[Hh];
    if (e < E_) {
        s = ei[e]; d = ei[E_ + e];
#pragma unroll
        for (int h = 0; h < Hh; ++h) a[h] = ae[(size_t)e * Hh + h];
    } else {
        s = d = e - E_;
#pragma unroll
        for (int h = 0; h < Hh; ++h) a[h] = loopc[h];
    }
#pragma unroll
    for (int h = 0; h < Hh; ++h) {
        float al = asrc[s * Hh + h] + adst[d * Hh + h] + a[h];
        al = al > 0.0f ? al : 0.2f * al;
        atomicMax(&amax[d * Hh + h], fenc(al));
    }
}

// ---------- pass B: fused exp-weight, denom, weighted message scatter ----------
// 32 lanes per edge; lane t handles channel h*32+t for each head h (coalesced atomics).
__global__ void edge_passB(const int* __restrict__ ei,
                           const float* __restrict__ ae, const float* __restrict__ loopc,
                           const float* __restrict__ asrc, const float* __restrict__ adst,
                           const unsigned* __restrict__ amax,
                           const float* __restrict__ xw,
                           float* __restrict__ denom, float* __restrict__ acc,
                           int E_, int N_) {
    long long gid = (long long)blockIdx.x * blockDim.x + threadIdx.x;
    int lane = (int)(gid & 31);
    long long e = gid >> 5;
    if (e >= (long long)E_ + N_) return;
    int s, d; float a[Hh];
    if (e < E_) {
        s = ei[e]; d = ei[E_ + e];
#pragma unroll
        for (int h = 0; h < Hh; ++h) a[h] = ae[e * Hh + h];
    } else {
        s = d = (int)(e - E_);
#pragma unroll
        for (int h = 0; h < Hh; ++h) a[h] = loopc[h];
    }
    float w[Hh];
#pragma unroll
    for (int h = 0; h < Hh; ++h) {
        float al = asrc[s * Hh + h] + adst[d * Hh + h] + a[h];
        al = al > 0.0f ? al : 0.2f * al;
        w[h] = __expf(al - fdec(amax[d * Hh + h]));
    }
    if (lane == 0) {
#pragma unroll
        for (int h = 0; h < Hh; ++h) atomicAdd(&denom[d * Hh + h], w[h]);
    }
    const float* xs = xw + (size_t)s * HC;
    float* od = acc + (size_t)d * HC;
#pragma unroll
    for (int h = 0; h < Hh; ++h) {
        int c = h * Cc + lane;
        atomicAdd(&od[c], w[h] * xs[c]);
    }
}

// ---------- finalize layer1: normalize + bias + relu (concat) ----------
__global__ void finalize_concat(const float* __restrict__ acc, const float* __restrict__ denom,
                                const float* __restrict__ bias, float* __restrict__ out, int n) {
    long long i = (long long)blockIdx.x * blockDim.x + threadIdx.x;
    if (i >= (long long)n * HC) return;
    int node = (int)(i >> 7), hc = (int)(i & 127), h = hc >> 5;
    float v = acc[i] / (denom[node * Hh + h] + 1e-16f) + bias[hc];
    out[i] = v > 0.0f ? v : 0.0f;
}

// ---------- finalize layer2: normalize, mean over heads, + bias ----------
__global__ void finalize_mean(const float* __restrict__ acc, const float* __restrict__ denom,
                              const float* __restrict__ bias, float* __restrict__ out, int n) {
    int i = blockIdx.x * blockDim.x + threadIdx.x;
    if (i >= n * Cc) return;
    int node = i >> 5, c = i & 31;
    float s = 0.0f;
#pragma unroll
    for (int h = 0; h < Hh; ++h)
        s += acc[(size_t)node * HC + h * Cc + c] / (denom[node * Hh + h] + 1e-16f);
    out[i] = s * 0.25f + bias[c];
}

// ---------- global mean pool (sums + counts) ----------
__global__ void pool_sum(const float* __restrict__ h2, const int* __restrict__ batch,
                         float* __restrict__ psum, float* __restrict__ pcnt, int n) {
    int i = blockIdx.x * blockDim.x + threadIdx.x;
    if (i >= n * Cc) return;
    int node = i >> 5, c = i & 31;
    int g = batch[node];
    atomicAdd(&psum[g * Cc + c], h2[i]);
    if (c == 0) atomicAdd(&pcnt[g], 1.0f);
}

// ---------- final MLP: [G,32] -> relu(@W3+b3) -> @W4+b4 -> [G,2] ----------
__global__ void graph_mlp(const float* __restrict__ psum, const float* __restrict__ pcnt,
                          const float* __restrict__ W3, const float* __restrict__ b3,
                          const float* __restrict__ W4, const float* __restrict__ b4,
                          float* __restrict__ out, int G_) {
    int g = blockIdx.x * blockDim.x + threadIdx.x;
    if (g >= G_) return;
    float cnt = pcnt[g]; cnt = cnt > 1.0f ? cnt : 1.0f;
    float p[Cc];
#pragma unroll
    for (int c = 0; c < Cc; ++c) p[c] = psum[g * Cc + c] / cnt;
    float o0 = b4[0], o1 = b4[1];
    for (int j = 0; j < Cc; ++j) {
        float t = b3[j];
        for (int c = 0; c < Cc; ++c) t += p[c] * W3[c * Cc + j];
        t = t > 0.0f ? t : 0.0f;
        o0 += t * W4[j * 2 + 0];
        o1 += t * W4[j * 2 + 1];
    }
    out[g * 2 + 0] = o0;
    out[g * 2 + 1] = o1;
}

extern "C" void kernel_launch(void* const* d_in, const int* in_sizes, int n_in,
                              void* d_out, int out_size, void* d_ws, size_t ws_size,
                              hipStream_t stream) {
    const float* x    = (const float*)d_in[0];
    const int*   ei   = (const int*)d_in[1];
    const float* ea   = (const float*)d_in[2];
    const int*   bat  = (const int*)d_in[3];
    const float* W1   = (const float*)d_in[4];
    const float* as1  = (const float*)d_in[5];
    const float* ad1  = (const float*)d_in[6];
    const float* We1  = (const float*)d_in[7];
    const float* ate1 = (const float*)d_in[8];
    const float* b1   = (const float*)d_in[9];
    const float* W2   = (const float*)d_in[10];
    const float* as2  = (const float*)d_in[11];
    const float* ad2  = (const float*)d_in[12];
    const float* We2  = (const float*)d_in[13];
    const float* ate2 = (const float*)d_in[14];
    const float* b2   = (const float*)d_in[15];
    const float* W3   = (const float*)d_in[16];
    const float* b3   = (const float*)d_in[17];
    const float* W4   = (const float*)d_in[18];
    const float* b4   = (const float*)d_in[19];
    float* out = (float*)d_out;

    const int N  = in_sizes[0] / HC;          // in = 128
    const int E  = in_sizes[1] / 2;
    const int G  = out_size / 2;

    // ---- workspace carve (floats) ----
    float* ws = (float*)d_ws;
    size_t o = 0;
    float*    xw    = ws + o; o += (size_t)N * HC;
    float*    acc   = ws + o; o += (size_t)N * HC;   // zero-region start
    float*    denom = ws + o; o += (size_t)N * Hh;
    unsigned* amax  = (unsigned*)(ws + o); o += (size_t)N * Hh;  // zero-region end
    float*    h1    = ws + o; o += (size_t)N * HC;
    float*    h2    = ws + o; o += (size_t)N * Cc;
    float*    asrc  = ws + o; o += (size_t)N * Hh;
    float*    adst  = ws + o; o += (size_t)N * Hh;
    float*    ae1   = ws + o; o += (size_t)E * Hh;
    float*    ae2   = ws + o; o += (size_t)E * Hh;
    float*    acc16 = ws + o; o += 16;               // misc zero-region start
    float*    M1    = ws + o; o += ED * Hh;
    float*    M2    = ws + o; o += ED * Hh;
    float*    L1    = ws + o; o += Hh;
    float*    L2    = ws + o; o += Hh;
    float*    psum  = ws + o; o += (size_t)G * Cc;
    float*    pcnt  = ws + o; o += (size_t)G;        // misc zero-region end
    (void)ws_size; (void)n_in;

    const int TB = 256;
    const long long zlen  = (long long)N * (HC + 2 * Hh);        // acc+denom+amax
    const long long mlen  = 16 + 2 * ED * Hh + 2 * Hh + (long long)G * (Cc + 1);
    const int tot = E + N;

    // ---- preprocessing: edge-attr mean, folded W_e, per-edge logits ----
    zero_f32<<<(int)((mlen + TB - 1) / TB), TB, 0, stream>>>(acc16, mlen);
    edge_attr_sum<<<512, TB, 0, stream>>>(ea, acc16, E);
    fold_We<<<1, 64, 0, stream>>>(We1, ate1, We2, ate2, M1, M2);
    loop_consts<<<1, Hh, 0, stream>>>(acc16, M1, M2, L1, L2, 1.0f / (float)E);
    edge_logits<<<(E + TB - 1) / TB, TB, 0, stream>>>(ea, M1, M2, ae1, ae2, E);

    // ================= layer 1 =================
    gemm128_wmma<<<(N + 127) / 128, TB, 0, stream>>>(x, W1, xw, N);
    att_dots<<<(N * Hh + TB - 1) / TB, TB, 0, stream>>>(xw, as1, ad1, asrc, adst, N);
    zero_f32<<<(int)((zlen + TB - 1) / TB), TB, 0, stream>>>(acc, zlen);
    edge_passA<<<(tot + TB - 1) / TB, TB, 0, stream>>>(ei, ae1, L1, asrc, adst, amax, E, N);
    {
        long long th = (long long)tot * 32;
        edge_passB<<<(int)((th + TB - 1) / TB), TB, 0, stream>>>(
            ei, ae1, L1, asrc, adst, amax, xw, denom, acc, E, N);
    }
    {
        long long th = (long long)N * HC;
        finalize_concat<<<(int)((th + TB - 1) / TB), TB, 0, stream>>>(acc, denom, b1, h1, N);
    }

    // ================= layer 2 =================
    gemm128_wmma<<<(N + 127) / 128, TB, 0, stream>>>(h1, W2, xw, N);
    att_dots<<<(N * Hh + TB - 1) / TB, TB, 0, stream>>>(xw, as2, ad2, asrc, adst, N);
    zero_f32<<<(int)((zlen + TB - 1) / TB), TB, 0, stream>>>(acc, zlen);
    edge_passA<<<(tot + TB - 1) / TB, TB, 0, stream>>>(ei, ae2, L2, asrc, adst, amax, E, N);
    {
        long long th = (long long)tot * 32;
        edge_passB<<<(int)((th + TB - 1) / TB), TB, 0, stream>>>(
            ei, ae2, L2, asrc, adst, amax, xw, denom, acc, E, N);
    }
    finalize_mean<<<(N * Cc + TB - 1) / TB, TB, 0, stream>>>(acc, denom, b2, h2, N);

    // ================= pool + MLP =================
    pool_sum<<<(N * Cc + TB - 1) / TB, TB, 0, stream>>>(h2, bat, psum, pcnt, N);
    graph_mlp<<<(G + 63) / 64, 64, 0, stream>>>(psum, pcnt, W3, b3, W4, b4, out, G);
}